// RadiotherapyLoss_66949950210778
// MI455X (gfx1250) — compile-verified
//
#include <hip/hip_runtime.h>

// ---------------------------------------------------------------------------
// Radiotherapy loss for MI455X (gfx1250): bandwidth-bound fused reduction.
// Stage 1: stream 201MB once (NT b128 loads), 24 per-thread accumulators,
//          block reduction via V_WMMA_F32_16X16X4_F32 column sums.
// Stage 2: deterministic fixed-order finalize (1 block).
// ---------------------------------------------------------------------------

typedef __attribute__((ext_vector_type(2))) float v2f;
typedef __attribute__((ext_vector_type(4))) float v4f;
typedef __attribute__((ext_vector_type(8))) float v8f;
typedef __attribute__((ext_vector_type(4))) int   v4i;

#define N_PTV 3
#define BATCH 4
#define DHW (128 * 128 * 128)            /* 2,097,152 voxels per batch item */
#define F4_PER_BATCH (DHW / 4)           /* 524,288 float4 per batch item   */
#define BLOCKS_PER_BATCH 256
#define F4_PER_BLOCK (F4_PER_BATCH / BLOCKS_PER_BATCH) /* 2048 */
#define THREADS 256
#define ITERS (F4_PER_BLOCK / THREADS)   /* 8 */
#define NACC 24                          /* live accumulators per thread */
#define NSLOT 32                         /* padded to 2 WMMA column panels */

// Accumulator slot layout:
//  0        : sum |output-target|
//  1..3     : msum[k]            (PTV label counts)
//  4..6     : sigmoid numerator, dvh_pred
//  7..9     : sigmoid numerator, dvh_true
//  10       : vox OAR   (sum OAR_mask)
//  11       : vox PTV   (sum float(PTV_mask))
//  12..14   : sum (pred*OAR)^{1,2,10}
//  15..17   : sum (pred*PTVf)^{1,2,10}
//  18..20   : sum (true*OAR)^{1,2,10}
//  21..23   : sum (true*PTVf)^{1,2,10}

__device__ __forceinline__ void accum_elem(float o, float tg, float p, float tr,
                                           float om, int lab,
                                           float t0, float t1, float t2,
                                           float* acc)
{
    acc[0] += fabsf(o - tg);
    float pf = (float)lab;
    acc[10] += om;
    acc[11] += pf;
    if (lab > 0) {                       // exactly one label matches per voxel
        int kk = lab - 1;
        float th = (kk == 0) ? t0 : ((kk == 1) ? t1 : t2);
        acc[1 + kk] += 1.0f;
        // sigmoid((dose - th)/0.1) = 1/(1+exp((th-dose)*10))
        acc[4 + kk] += 1.0f / (1.0f + __expf((th - p)  * 10.0f));
        acc[7 + kk] += 1.0f / (1.0f + __expf((th - tr) * 10.0f));
    }
    float sd, sd2, sd4;
    sd = p * om;  sd2 = sd * sd; sd4 = sd2 * sd2;
    acc[12] += sd; acc[13] += sd2; acc[14] += sd4 * sd4 * sd2;
    sd = p * pf;  sd2 = sd * sd; sd4 = sd2 * sd2;
    acc[15] += sd; acc[16] += sd2; acc[17] += sd4 * sd4 * sd2;
    sd = tr * om; sd2 = sd * sd; sd4 = sd2 * sd2;
    acc[18] += sd; acc[19] += sd2; acc[20] += sd4 * sd4 * sd2;
    sd = tr * pf; sd2 = sd * sd; sd4 = sd2 * sd2;
    acc[21] += sd; acc[22] += sd2; acc[23] += sd4 * sd4 * sd2;
}

__global__ __launch_bounds__(THREADS)
void rt_loss_stage1(const v4f* __restrict__ outp, const v4f* __restrict__ tgt,
                    const v4f* __restrict__ dvp,  const v4f* __restrict__ dvt,
                    const v4f* __restrict__ oar,  const v4i* __restrict__ ptv,
                    const float* __restrict__ pv, float* __restrict__ partials)
{
    const int bl   = blockIdx.x;
    const int b    = bl / BLOCKS_PER_BATCH;
    const int blk  = bl % BLOCKS_PER_BATCH;
    const int t    = threadIdx.x;
    const int base = b * F4_PER_BATCH + blk * F4_PER_BLOCK;

    const float t0 = pv[0], t1 = pv[1], t2 = pv[2];

    float acc[NACC];
#pragma unroll
    for (int a = 0; a < NACC; ++a) acc[a] = 0.0f;

#pragma unroll
    for (int j = 0; j < ITERS; ++j) {
        const int idx = base + j * THREADS + t;
        // Single-use stream: non-temporal so 201MB doesn't churn L2.
        v4f o  = __builtin_nontemporal_load(&outp[idx]);
        v4f tg = __builtin_nontemporal_load(&tgt[idx]);
        v4f p  = __builtin_nontemporal_load(&dvp[idx]);
        v4f tr = __builtin_nontemporal_load(&dvt[idx]);
        v4f om = __builtin_nontemporal_load(&oar[idx]);
        v4i lb = __builtin_nontemporal_load(&ptv[idx]);
        accum_elem(o.x, tg.x, p.x, tr.x, om.x, lb.x, t0, t1, t2, acc);
        accum_elem(o.y, tg.y, p.y, tr.y, om.y, lb.y, t0, t1, t2, acc);
        accum_elem(o.z, tg.z, p.z, tr.z, om.z, lb.z, t0, t1, t2, acc);
        accum_elem(o.w, tg.w, p.w, tr.w, om.w, lb.w, t0, t1, t2, acc);
    }

    // ---- Block reduction: column-sum a 256x32 LDS matrix with WMMA ----
    __shared__ float X[THREADS][NSLOT + 1];   // +1 pad vs bank conflicts
#pragma unroll
    for (int a = 0; a < NACC; ++a) X[t][a] = acc[a];
#pragma unroll
    for (int a = NACC; a < NSLOT; ++a) X[t][a] = 0.0f;
    __syncthreads();

    if (t < 32) {                 // wave 0, EXEC all-ones as WMMA requires
        v2f ones = {1.0f, 1.0f};  // A = ones(16x4): D[m][n] = sum_k B[k][n] + C
        v8f c0 = {};              // panel cols 0..15
        v8f c1 = {};              // panel cols 16..31
        const int col  = t & 15;
        const int roff = (t >= 16) ? 2 : 0;   // B 4x16: lanes 16-31 carry K=2,3
        for (int r = 0; r < THREADS / 4; ++r) {
            const int r0 = 4 * r + roff;
            v2f b0, b1;
            b0.x = X[r0][col];        b0.y = X[r0 + 1][col];
            b1.x = X[r0][col + 16];   b1.y = X[r0 + 1][col + 16];
            c0 = __builtin_amdgcn_wmma_f32_16x16x4_f32(
                     false, ones, false, b0, (short)0, c0, false, false);
            c1 = __builtin_amdgcn_wmma_f32_16x16x4_f32(
                     false, ones, false, b1, (short)0, c1, false, false);
        }
        // D row 0, lane n<16 holds colsum[n] (rows identical since A is ones).
        if (t < 16) {
            partials[bl * NSLOT + t]      = c0[0];
            partials[bl * NSLOT + 16 + t] = c1[0];
        }
    }
}

__global__ __launch_bounds__(128)
void rt_loss_stage2(const float* __restrict__ partials, float* __restrict__ out)
{
    __shared__ float S[BATCH][NSLOT];
    const int t = threadIdx.x;            // 128 = 4 batches * 32 slots
    const int b = t >> 5;
    const int a = t & 31;
    double s = 0.0;
    for (int i = 0; i < BLOCKS_PER_BATCH; ++i)
        s += (double)partials[(b * BLOCKS_PER_BATCH + i) * NSLOT + a];
    S[b][a] = (float)s;
    __syncthreads();

    if (t == 0) {
        // MAE over all voxels
        double mae = 0.0;
        for (int bb = 0; bb < BATCH; ++bb) mae += S[bb][0];
        mae /= (double)BATCH * (double)DHW;

        // DVH loss: mean over (k,b) of (num_p/msum - num_t/msum)^2, 0 if msum==0
        double dvh = 0.0;
        for (int bb = 0; bb < BATCH; ++bb)
            for (int k = 0; k < N_PTV; ++k) {
                double ms = S[bb][1 + k];
                double dp = (ms > 0.0) ? (double)S[bb][4 + k] / ms : 0.0;
                double dt = (ms > 0.0) ? (double)S[bb][7 + k] / ms : 0.0;
                double d = dp - dt;
                dvh += d * d;
            }
        dvh /= (double)(BATCH * N_PTV);

        // Moment loss: sum_p sum_{struct,b} ((Sp/vox)^(1/p) - (St/vox)^(1/p))^2 / B
        double mom = 0.0;
        for (int bb = 0; bb < BATCH; ++bb)
            for (int si = 0; si < 2; ++si) {
                double vx = S[bb][10 + si];
                for (int pi = 0; pi < 3; ++pi) {
                    double mp = (double)S[bb][12 + si * 3 + pi] / vx;
                    double mt = (double)S[bb][18 + si * 3 + pi] / vx;
                    if (pi == 1)      { mp = sqrt(mp);      mt = sqrt(mt); }
                    else if (pi == 2) { mp = pow(mp, 0.1);  mt = pow(mt, 0.1); }
                    double d = mp - mt;
                    mom += d * d;
                }
            }
        mom /= (double)BATCH;

        out[0] = (float)(0.5 * mae + 0.05 * dvh + 0.1 * mom);
    }
}

extern "C" void kernel_launch(void* const* d_in, const int* in_sizes, int n_in,
                              void* d_out, int out_size, void* d_ws, size_t ws_size,
                              hipStream_t stream)
{
    (void)in_sizes; (void)n_in; (void)out_size; (void)ws_size;
    const v4f*  outp = (const v4f*)d_in[0];
    const v4f*  tgt  = (const v4f*)d_in[1];
    const v4f*  dvp  = (const v4f*)d_in[2];
    const v4f*  dvt  = (const v4f*)d_in[3];
    const v4f*  oar  = (const v4f*)d_in[4];
    const v4i*  ptv  = (const v4i*)d_in[5];
    const float* pv  = (const float*)d_in[6];

    float* partials = (float*)d_ws;   // 1024 blocks * 32 floats = 128 KB

    rt_loss_stage1<<<BATCH * BLOCKS_PER_BATCH, THREADS, 0, stream>>>(
        outp, tgt, dvp, dvt, oar, ptv, pv, partials);
    rt_loss_stage2<<<1, 128, 0, stream>>>(partials, (float*)d_out);
}